// Attention_8194797600790
// MI455X (gfx1250) — compile-verified
//
#include <hip/hip_runtime.h>
#include <math.h>

// ---------------- problem constants ----------------
#define BATCH 2
#define SEQ   4096          // N
#define DIMC  256           // C
#define HEADS 8
#define DH    32            // C / HEADS
#define MRED  1024          // (H/SR)*(W/SR) = 32*32
#define IMW   64            // W = H = 64

typedef float v2f __attribute__((ext_vector_type(2)));
typedef float v8f __attribute__((ext_vector_type(8)));

// fp32 tensor-core MAC: D(16x16) = A(16x4) * B(4x16) + C
__device__ __forceinline__ v8f wmma4(v2f a, v2f b, v8f c) {
    return __builtin_amdgcn_wmma_f32_16x16x4_f32(
        /*neg_a*/false, a, /*neg_b*/false, b,
        /*c_mod*/(short)0, c, /*reuse_a*/false, /*reuse_b*/false);
}

__device__ __forceinline__ float half16_max(float v) {
    #pragma unroll
    for (int off = 1; off < 16; off <<= 1)
        v = fmaxf(v, __shfl_xor(v, off, 32));
    return v;
}
__device__ __forceinline__ float half16_sum(float v) {
    #pragma unroll
    for (int off = 1; off < 16; off <<= 1)
        v += __shfl_xor(v, off, 32);
    return v;
}

// ---------------------------------------------------------------------------
// K0: fold LoRA into weights.
//   wq_eff = q_w + b_q @ a_q          (delta_q = x @ a_q.T @ b_q.T = x @ (b_q a_q).T)
//   wk_eff = kv_w[0:256]   + b_v @ a_v
//   wv_eff = kv_w[256:512] + b_v @ a_v
// ---------------------------------------------------------------------------
__global__ void prep_weff_kernel(const float* __restrict__ q_w,
                                 const float* __restrict__ b_q,
                                 const float* __restrict__ a_q,
                                 const float* __restrict__ kv_w,
                                 const float* __restrict__ b_v,
                                 const float* __restrict__ a_v,
                                 float* __restrict__ wq,
                                 float* __restrict__ wk,
                                 float* __restrict__ wv) {
    int n = blockIdx.x;          // output-feature row of W (256)
    int k = threadIdx.x;         // input-feature col (256)
    float accq = 0.f, accv = 0.f;
    #pragma unroll
    for (int r = 0; r < 8; ++r) {
        accq += b_q[n * 8 + r] * a_q[r * DIMC + k];
        accv += b_v[n * 8 + r] * a_v[r * DIMC + k];
    }
    wq[n * DIMC + k] = q_w[n * DIMC + k] + accq;
    wk[n * DIMC + k] = kv_w[n * DIMC + k] + accv;
    wv[n * DIMC + k] = kv_w[(n + DIMC) * DIMC + k] + accv;
}

// ---------------------------------------------------------------------------
// K1/K5: Out[rows,256] = A[rows,256] @ W[256,256].T + bias
// one wave per 16x16 output tile; 64 fp32-WMMA k-steps.
// grid = (rows/16, 256/16), block = 32
// ---------------------------------------------------------------------------
__global__ __launch_bounds__(32)
void gemm_bias_kernel(const float* __restrict__ A,
                      const float* __restrict__ W,
                      const float* __restrict__ bias,
                      float* __restrict__ Out) {
    const int lane = threadIdx.x;
    const int m    = lane & 15;
    const int half = lane >> 4;
    const int row0 = blockIdx.x * 16;
    const int col0 = blockIdx.y * 16;

    const float* arow = A + (size_t)(row0 + m) * DIMC;   // A-frag: row m
    const float* wrow = W + (size_t)(col0 + m) * DIMC;   // B-frag: B[k][n]=W[n][k]

    v8f c = {};
    #pragma unroll 4
    for (int k0 = 0; k0 < DIMC; k0 += 4) {
        v2f a = *(const v2f*)(arow + k0 + 2 * half);
        v2f b = *(const v2f*)(wrow + k0 + 2 * half);
        c = wmma4(a, b, c);
    }
    const float bv = bias[col0 + m];
    #pragma unroll
    for (int r = 0; r < 8; ++r) {
        int row = row0 + r + 8 * half;
        Out[(size_t)row * DIMC + col0 + m] = c[r] + bv;
    }
}

// ---------------------------------------------------------------------------
// K2: strided 2x2 conv as 4 accumulated GEMMs.
//   xs_pre[b, oh*32+ow, co] = sum_{i,j,ci} x[b, (2oh+i)*64 + 2ow+j, ci] * sr_w[co,ci,i,j] + sr_b
// grid = (2048/16, 256/16), block = 32
// ---------------------------------------------------------------------------
__global__ __launch_bounds__(32)
void conv_sr_kernel(const float* __restrict__ x,
                    const float* __restrict__ sr_w,
                    const float* __restrict__ sr_b,
                    float* __restrict__ xs_pre) {
    const int lane = threadIdx.x;
    const int m    = lane & 15;
    const int half = lane >> 4;
    const int row0 = blockIdx.x * 16;   // reduced-pixel rows (B*MRED total)
    const int col0 = blockIdx.y * 16;   // output channels

    const int mrow = row0 + m;          // this lane's A row (reduced pixel)
    const int b    = mrow >> 10;        // / MRED
    const int ml   = mrow & (MRED - 1);
    const int oh   = ml >> 5;
    const int ow   = ml & 31;
    const int co   = col0 + m;          // this lane's B col (out channel)

    v8f c = {};
    #pragma unroll
    for (int i = 0; i < 2; ++i) {
        #pragma unroll
        for (int j = 0; j < 2; ++j) {
            const int n_pix = (2 * oh + i) * IMW + 2 * ow + j;
            const float* ar = x + ((size_t)b * SEQ + n_pix) * DIMC;
            for (int k0 = 0; k0 < DIMC; k0 += 4) {
                v2f a = *(const v2f*)(ar + k0 + 2 * half);
                int kk = k0 + 2 * half;               // ci
                v2f bf;
                bf.x = sr_w[((co * DIMC + kk) * 2 + i) * 2 + j];
                bf.y = sr_w[((co * DIMC + kk + 1) * 2 + i) * 2 + j];
                c = wmma4(a, bf, c);
            }
        }
    }
    const float bv = sr_b[co];
    #pragma unroll
    for (int r = 0; r < 8; ++r) {
        int row = row0 + r + 8 * half;
        xs_pre[(size_t)row * DIMC + col0 + m] = c[r] + bv;
    }
}

// ---------------------------------------------------------------------------
// K3: LayerNorm over channel dim (256). grid = B*MRED rows, block = 256.
// biased variance (ddof=0) to match jnp.var.
// ---------------------------------------------------------------------------
__global__ __launch_bounds__(256)
void layernorm_kernel(const float* __restrict__ in,
                      const float* __restrict__ g,
                      const float* __restrict__ bta,
                      float* __restrict__ out) {
    __shared__ float sm[DIMC];
    const int row = blockIdx.x;
    const int tid = threadIdx.x;
    const float v = in[(size_t)row * DIMC + tid];

    sm[tid] = v; __syncthreads();
    for (int s = 128; s > 0; s >>= 1) {
        if (tid < s) sm[tid] += sm[tid + s];
        __syncthreads();
    }
    const float mu = sm[0] * (1.f / DIMC);
    __syncthreads();
    const float d = v - mu;
    sm[tid] = d * d; __syncthreads();
    for (int s = 128; s > 0; s >>= 1) {
        if (tid < s) sm[tid] += sm[tid + s];
        __syncthreads();
    }
    const float var = sm[0] * (1.f / DIMC);
    const float rstd = rsqrtf(var + 1e-5f);
    out[(size_t)row * DIMC + tid] = d * rstd * g[tid] + bta[tid];
}

// ---------------------------------------------------------------------------
// K4: fused flash attention per (b, head, 16-query tile).
//   S = (q k^T)*scale via WMMA (dh=32 -> 8 k-steps), online softmax in
//   C-layout registers (row stats via 16-lane xor-shuffles), P transposed
//   through 1KB LDS into A-layout, O += P@V via WMMA (2 col-groups x 4 steps).
// grid = (SEQ/16, HEADS, BATCH), block = 32 (one wave).
// ---------------------------------------------------------------------------
__global__ __launch_bounds__(32)
void attention_kernel(const float* __restrict__ Q,
                      const float* __restrict__ Km,
                      const float* __restrict__ Vm,
                      float* __restrict__ O) {
    __shared__ float pl[16 * 16];

    const int lane = threadIdx.x;
    const int m    = lane & 15;
    const int half = lane >> 4;
    const int q0   = blockIdx.x * 16;
    const int h    = blockIdx.y;
    const int b    = blockIdx.z;
    const float scale = 0.1767766952966369f;   // 1/sqrt(32)

    // preload scaled Q tile as 8 A-fragments (K = dh = 32)
    const float* qb = Q + ((size_t)(b * SEQ + q0 + m)) * DIMC + h * DH;
    v2f aq[8];
    #pragma unroll
    for (int s = 0; s < 8; ++s) {
        v2f t = *(const v2f*)(qb + 4 * s + 2 * half);
        aq[s].x = t.x * scale;
        aq[s].y = t.y * scale;
    }

    float mrow[8], lrow[8];
    #pragma unroll
    for (int r = 0; r < 8; ++r) { mrow[r] = -3.402823466e38f; lrow[r] = 0.f; }
    v8f o0 = {}, o1 = {};

    for (int kt = 0; kt < MRED / 16; ++kt) {
        const int k0 = kt * 16;
        // ---- S = q k^T (scaled) ----
        const float* kb = Km + ((size_t)(b * MRED + k0 + m)) * DIMC + h * DH;
        v8f s = {};
        #pragma unroll
        for (int st = 0; st < 8; ++st) {
            v2f bf = *(const v2f*)(kb + 4 * st + 2 * half);
            s = wmma4(aq[st], bf, s);
        }
        // ---- online softmax (C-layout; rows r+8*half across 16 lanes) ----
        #pragma unroll
        for (int r = 0; r < 8; ++r) {
            float sv   = s[r];
            float mx   = half16_max(sv);
            float mnew = fmaxf(mrow[r], mx);
            float p    = expf(sv - mnew);
            float rsum = half16_sum(p);
            float alpha = expf(mrow[r] - mnew);
            lrow[r] = lrow[r] * alpha + rsum;
            mrow[r] = mnew;
            o0[r] *= alpha;
            o1[r] *= alpha;
            s[r] = p;
        }
        // ---- transpose P (C-layout -> A-layout) through LDS ----
        __syncthreads();
        #pragma unroll
        for (int r = 0; r < 8; ++r)
            pl[(r + 8 * half) * 16 + m] = s[r];
        __syncthreads();
        // ---- O += P @ V ----
        const float* vb = Vm + ((size_t)(b * MRED + k0)) * DIMC + h * DH;
        #pragma unroll
        for (int st = 0; st < 4; ++st) {
            v2f ap;
            ap.x = pl[m * 16 + 4 * st + 2 * half];
            ap.y = pl[m * 16 + 4 * st + 2 * half + 1];
            const int kk = 4 * st + 2 * half;
            v2f b0, b1;
            b0.x = vb[(size_t)kk * DIMC + m];            // d cols 0..15
            b0.y = vb[(size_t)(kk + 1) * DIMC + m];
            b1.x = vb[(size_t)kk * DIMC + 16 + m];       // d cols 16..31
            b1.y = vb[(size_t)(kk + 1) * DIMC + 16 + m];
            o0 = wmma4(ap, b0, o0);
            o1 = wmma4(ap, b1, o1);
        }
    }
    // ---- normalize & write [B,N,C] with c = h*32 + d ----
    #pragma unroll
    for (int r = 0; r < 8; ++r) {
        const float inv = 1.f / lrow[r];
        const int row = q0 + r + 8 * half;
        float* op = O + ((size_t)(b * SEQ + row)) * DIMC + h * DH;
        op[m]      = o0[r] * inv;
        op[16 + m] = o1[r] * inv;
    }
}

// ---------------------------------------------------------------------------
extern "C" void kernel_launch(void* const* d_in, const int* in_sizes, int n_in,
                              void* d_out, int out_size, void* d_ws, size_t ws_size,
                              hipStream_t stream) {
    const float* x      = (const float*)d_in[0];
    const float* q_w    = (const float*)d_in[1];
    const float* q_b    = (const float*)d_in[2];
    const float* kv_w   = (const float*)d_in[3];
    const float* kv_b   = (const float*)d_in[4];
    const float* proj_w = (const float*)d_in[5];
    const float* proj_b = (const float*)d_in[6];
    const float* a_q    = (const float*)d_in[7];
    const float* b_q    = (const float*)d_in[8];
    const float* a_v    = (const float*)d_in[9];
    const float* b_v    = (const float*)d_in[10];
    const float* sr_w   = (const float*)d_in[11];
    const float* sr_b   = (const float*)d_in[12];
    const float* ln_g   = (const float*)d_in[13];
    const float* ln_b   = (const float*)d_in[14];
    float* out = (float*)d_out;

    // workspace layout (floats)
    float* ws = (float*)d_ws;
    float* wq_eff = ws;                         // 256*256
    float* wk_eff = wq_eff + DIMC * DIMC;
    float* wv_eff = wk_eff + DIMC * DIMC;
    float* Qbuf   = wv_eff + DIMC * DIMC;       // 8192*256
    float* xs_pre = Qbuf   + (size_t)BATCH * SEQ * DIMC;   // 2048*256
    float* xs     = xs_pre + (size_t)BATCH * MRED * DIMC;
    float* Kmat   = xs     + (size_t)BATCH * MRED * DIMC;
    float* Vmat   = Kmat   + (size_t)BATCH * MRED * DIMC;
    float* AO     = Vmat   + (size_t)BATCH * MRED * DIMC;  // 8192*256

    // K0: fold LoRA into projection weights
    prep_weff_kernel<<<dim3(DIMC), dim3(DIMC), 0, stream>>>(
        q_w, b_q, a_q, kv_w, b_v, a_v, wq_eff, wk_eff, wv_eff);

    // K1: Q = x @ Wq_eff.T + q_b
    gemm_bias_kernel<<<dim3(BATCH * SEQ / 16, DIMC / 16), dim3(32), 0, stream>>>(
        x, wq_eff, q_b, Qbuf);

    // K2: spatial reduction conv
    conv_sr_kernel<<<dim3(BATCH * MRED / 16, DIMC / 16), dim3(32), 0, stream>>>(
        x, sr_w, sr_b, xs_pre);

    // K3: LayerNorm
    layernorm_kernel<<<dim3(BATCH * MRED), dim3(DIMC), 0, stream>>>(
        xs_pre, ln_g, ln_b, xs);

    // K1': K = xs @ Wk_eff.T + kv_b[0:256] ; V = xs @ Wv_eff.T + kv_b[256:512]
    gemm_bias_kernel<<<dim3(BATCH * MRED / 16, DIMC / 16), dim3(32), 0, stream>>>(
        xs, wk_eff, kv_b, Kmat);
    gemm_bias_kernel<<<dim3(BATCH * MRED / 16, DIMC / 16), dim3(32), 0, stream>>>(
        xs, wv_eff, kv_b + DIMC, Vmat);

    // K4: fused flash attention
    attention_kernel<<<dim3(SEQ / 16, HEADS, BATCH), dim3(32), 0, stream>>>(
        Qbuf, Kmat, Vmat, AO);

    // K5: out = AO @ proj_w.T + proj_b
    gemm_bias_kernel<<<dim3(BATCH * SEQ / 16, DIMC / 16), dim3(32), 0, stream>>>(
        AO, proj_w, proj_b, out);
}